// Tensor_4277787427180
// MI455X (gfx1250) — compile-verified
//
#include <hip/hip_runtime.h>

// ---------------------------------------------------------------------------
// Compile-time real-basis Clebsch-Gordan tensor (matches the reference's
// Racah formula + complex->real unitary; odd l1+l2+l3 -> imaginary part).
// ---------------------------------------------------------------------------
namespace cgc {

constexpr double fact(int n) { double r = 1.0; for (int i = 2; i <= n; ++i) r *= (double)i; return r; }

constexpr double sqrt_nr(double x) {
  if (x <= 0.0) return 0.0;
  double g = (x >= 1.0) ? x : 1.0;
  for (int i = 0; i < 48; ++i) g = 0.5 * (g + x / g);
  return g;
}

constexpr int imax3(int a, int b, int c) { int m = a; if (b > m) m = b; if (c > m) m = c; return m; }
constexpr int imin3(int a, int b, int c) { int m = a; if (b < m) m = b; if (c < m) m = c; return m; }

constexpr double cg_complex(int l1, int m1, int l2, int m2, int l3, int m3) {
  if (m1 + m2 != m3) return 0.0;
  int d = l1 - l2; if (d < 0) d = -d;
  if (l3 < d || l3 > l1 + l2) return 0.0;
  double pref = (double)(2 * l3 + 1) * fact(l1 + l2 - l3) * fact(l1 - l2 + l3) *
                fact(-l1 + l2 + l3) / fact(l1 + l2 + l3 + 1);
  pref *= fact(l3 + m3) * fact(l3 - m3) * fact(l1 - m1) * fact(l1 + m1) *
          fact(l2 - m2) * fact(l2 + m2);
  double s = 0.0;
  const int kmin = imax3(0, l2 - l3 - m1, l1 - l3 + m2);
  const int kmax = imin3(l1 + l2 - l3, l1 - m1, l2 + m2);
  for (int k = kmin; k <= kmax; ++k) {
    const double den = fact(k) * fact(l1 + l2 - l3 - k) * fact(l1 - m1 - k) *
                       fact(l2 + m2 - k) * fact(l3 - l2 + m1 + k) * fact(l3 - l1 - m2 + k);
    s += ((k & 1) ? -1.0 : 1.0) / den;
  }
  return sqrt_nr(pref) * s;
}

struct Cx { double re, im; };
constexpr Cx cmul(Cx a, Cx b) { return Cx{a.re * b.re - a.im * b.im, a.re * b.im + a.im * b.re}; }

// U(l)[i][j]: rows = real index (m=-l..l), cols = complex m index.
constexpr Cx umat(int l, int i, int j, double s) {
  if (i == l) return (j == l) ? Cx{1.0, 0.0} : Cx{0.0, 0.0};
  if (i > l) {
    const int m = i - l;
    const double sg = (m & 1) ? -1.0 : 1.0;
    if (j == l + m) return Cx{sg * s, 0.0};
    if (j == l - m) return Cx{s, 0.0};
    return Cx{0.0, 0.0};
  }
  const int m = l - i;
  const double sg = (m & 1) ? -1.0 : 1.0;
  if (j == l - m) return Cx{0.0, s};
  if (j == l + m) return Cx{0.0, -sg * s};
  return Cx{0.0, 0.0};
}

struct Blk { float v[5][5][5]; };

constexpr Blk mkblk(int l1, int l2, int l3) {
  Blk B{};
  const double s = sqrt_nr(0.5);
  double Cc[5][5] = {};
  for (int a = 0; a < 2 * l1 + 1; ++a)
    for (int b = 0; b < 2 * l2 + 1; ++b) {
      const int m3 = (a - l1) + (b - l2);
      if (m3 < -l3 || m3 > l3) continue;
      Cc[a][b] = cg_complex(l1, a - l1, l2, b - l2, l3, m3);
    }
  const bool even = (((l1 + l2 + l3) & 1) == 0);
  for (int i = 0; i < 2 * l1 + 1; ++i)
    for (int j = 0; j < 2 * l2 + 1; ++j)
      for (int k = 0; k < 2 * l3 + 1; ++k) {
        Cx acc{0.0, 0.0};
        for (int a = 0; a < 2 * l1 + 1; ++a) {
          const Cx u1 = umat(l1, i, a, s);
          if (u1.re == 0.0 && u1.im == 0.0) continue;
          for (int b = 0; b < 2 * l2 + 1; ++b) {
            const double cc = Cc[a][b];
            if (cc == 0.0) continue;
            const int c = (a - l1) + (b - l2) + l3;
            const Cx u2 = umat(l2, j, b, s);
            if (u2.re == 0.0 && u2.im == 0.0) continue;
            Cx u3 = umat(l3, k, c, s);
            u3.im = -u3.im;  // conj
            if (u3.re == 0.0 && u3.im == 0.0) continue;
            const Cx p = cmul(cmul(u1, u2), u3);
            acc.re += cc * p.re;
            acc.im += cc * p.im;
          }
        }
        double val = even ? acc.re : acc.im;
        if (val < 1e-12 && val > -1e-12) val = 0.0;
        B.v[i][j][k] = (float)val;
      }
  return B;
}

// One constexpr evaluation per block keeps each under clang's constexpr-step limit.
constexpr Blk B0  = mkblk(0, 0, 0);
constexpr Blk B1  = mkblk(0, 1, 1);
constexpr Blk B2  = mkblk(0, 2, 2);
constexpr Blk B3  = mkblk(1, 0, 1);
constexpr Blk B4  = mkblk(1, 1, 0);
constexpr Blk B5  = mkblk(1, 1, 1);
constexpr Blk B6  = mkblk(1, 1, 2);
constexpr Blk B7  = mkblk(1, 2, 1);
constexpr Blk B8  = mkblk(1, 2, 2);
constexpr Blk B9  = mkblk(2, 0, 2);
constexpr Blk B10 = mkblk(2, 1, 1);
constexpr Blk B11 = mkblk(2, 1, 2);
constexpr Blk B12 = mkblk(2, 2, 0);
constexpr Blk B13 = mkblk(2, 2, 1);
constexpr Blk B14 = mkblk(2, 2, 2);

struct Tab { float v[9][9][9]; };

constexpr Tab assemble() {
  Tab t{};
  const Blk* bp[15] = {&B0, &B1, &B2, &B3, &B4, &B5, &B6, &B7, &B8, &B9, &B10, &B11, &B12, &B13, &B14};
  const int L1a[15] = {0, 0, 0, 1, 1, 1, 1, 1, 1, 2, 2, 2, 2, 2, 2};
  const int L2a[15] = {0, 1, 2, 0, 1, 1, 1, 2, 2, 0, 1, 1, 2, 2, 2};
  const int L3a[15] = {0, 1, 2, 1, 0, 1, 2, 1, 2, 2, 1, 2, 0, 1, 2};
  for (int kc = 0; kc < 15; ++kc) {
    const int l1 = L1a[kc], l2 = L2a[kc], l3 = L3a[kc];
    for (int i = 0; i < 2 * l1 + 1; ++i)
      for (int j = 0; j < 2 * l2 + 1; ++j)
        for (int k = 0; k < 2 * l3 + 1; ++k)
          t.v[l1 * l1 + i][l2 * l2 + j][l3 * l3 + k] = bp[kc]->v[i][j][k];
  }
  return t;
}

constexpr Tab CGT = assemble();

}  // namespace cgc

// Combo id for allowed (l1,l2,l3); -1 if triangle-forbidden.
__host__ __device__ constexpr int cid(int l1, int l2, int l3) {
  constexpr int T[3][3][3] = {
      {{0, -1, -1}, {-1, 1, -1}, {-1, -1, 2}},
      {{-1, 3, -1}, {4, 5, 6}, {-1, 7, 8}},
      {{-1, -1, 9}, {-1, 10, 11}, {12, 13, 14}}};
  return T[l1][l2][l3];
}

// Flat (l1,l2,r,l3) element offset into the kernel tensor for each allowed
// combo, with r = (l1+l2)&1 (the parity mask zeroes the other r channel).
// Compile-time: folds into immediates in the unrolled async staging.
constexpr int kWOFF[15] = {0, 10, 14, 22, 24, 25, 26, 34, 35, 38, 46, 47, 48, 49, 50};

// ---------------------------------------------------------------------------
// Kernel: blockDim = (128 f-lanes, 4 n-rows). Weights (15 x 128 floats) are
// broadcast-staged into LDS via gfx1250 async global->LDS copies (unrolled,
// immediate offsets, issued by the first 4 waves back-to-back); the 600 MB of
// x/y/out stream through NT-hinted loads/stores (preserve L2 for weights).
// ---------------------------------------------------------------------------
__global__ __launch_bounds__(512) void tp_kernel(const float* __restrict__ x1,
                                                 const float* __restrict__ x2,
                                                 const float* __restrict__ kern,
                                                 float* __restrict__ out, int N) {
  __shared__ float wlds[15 * 128];

  const int f = threadIdx.x;  // 0..127

  // --- stage weight table into LDS with async copies (ASYNCcnt path) ---
  // threadIdx.y==0 -> first 4 waves; each issues 15 async b32 copies whose
  // per-lane global offsets are f*4 + compile-time-constant immediates.
  if (threadIdx.y == 0) {
#pragma unroll
    for (int k = 0; k < 15; ++k) {
      const unsigned byte_off = (unsigned)((kWOFF[k] * 128 + f) * 4);
      const unsigned lds_off = (unsigned)(size_t)(&wlds[k * 128 + f]);
      asm volatile("global_load_async_to_lds_b32 %0, %1, %2"
                   :
                   : "v"(lds_off), "v"(byte_off), "s"(kern)
                   : "memory");
    }
  }
  asm volatile("s_wait_asynccnt 0" ::: "memory");
  __syncthreads();

  const int n = blockIdx.x * 4 + threadIdx.y;
  if (n >= N) return;

  const float* px = x1 + (size_t)n * 9 * 128 + f;
  const float* py = x2 + (size_t)n * 9 * 128 + f;

  float x[9], y[9];
#pragma unroll
  for (int a = 0; a < 9; ++a) x[a] = __builtin_nontemporal_load(px + a * 128);
#pragma unroll
  for (int b = 0; b < 9; ++b) y[b] = __builtin_nontemporal_load(py + b * 128);

  float w[15];
#pragma unroll
  for (int k = 0; k < 15; ++k) w[k] = wlds[k * 128 + f];

  float acc[2][9] = {{0.f, 0.f, 0.f, 0.f, 0.f, 0.f, 0.f, 0.f, 0.f},
                     {0.f, 0.f, 0.f, 0.f, 0.f, 0.f, 0.f, 0.f, 0.f}};

  // Fully unrolled; CG table reads constant-fold, zero paths are dead-coded.
#pragma unroll
  for (int l1 = 0; l1 < 3; ++l1) {
#pragma unroll
    for (int l2 = 0; l2 < 3; ++l2) {
#pragma unroll
      for (int l3 = 0; l3 < 3; ++l3) {
        const int kc = cid(l1, l2, l3);
        if (kc < 0) continue;
        const int r = (l1 + l2) & 1;
        const float wv = w[kc];
#pragma unroll
        for (int i = 0; i < 2 * l1 + 1; ++i) {
#pragma unroll
          for (int j = 0; j < 2 * l2 + 1; ++j) {
            const float t = x[l1 * l1 + i] * y[l2 * l2 + j] * wv;
#pragma unroll
            for (int k3 = 0; k3 < 2 * l3 + 1; ++k3) {
              const float cgv = cgc::CGT.v[l1 * l1 + i][l2 * l2 + j][l3 * l3 + k3];
              if (cgv != 0.0f)
                acc[r][l3 * l3 + k3] = __builtin_fmaf(cgv, t, acc[r][l3 * l3 + k3]);
            }
          }
        }
      }
    }
  }

  float* po = out + (size_t)n * 18 * 128 + f;
#pragma unroll
  for (int r = 0; r < 2; ++r)
#pragma unroll
    for (int c = 0; c < 9; ++c)
      __builtin_nontemporal_store(acc[r][c], po + (r * 9 + c) * 128);
}

extern "C" void kernel_launch(void* const* d_in, const int* in_sizes, int n_in,
                              void* d_out, int out_size, void* d_ws, size_t ws_size,
                              hipStream_t stream) {
  const float* x1 = (const float*)d_in[0];    // (N,1,9,128)
  const float* x2 = (const float*)d_in[1];    // (N,1,9,128)
  const float* kern = (const float*)d_in[2];  // (1,3,1,3,2,3,128)
  float* out = (float*)d_out;                 // (N,2,9,128)

  const int N = in_sizes[0] / (9 * 128);
  dim3 block(128, 4, 1);
  dim3 grid((N + 3) / 4, 1, 1);
  hipLaunchKernelGGL(tp_kernel, grid, block, 0, stream, x1, x2, kern, out, N);
}